// Layer_18554258719295
// MI455X (gfx1250) — compile-verified
//
#include <hip/hip_runtime.h>
#include <stdint.h>

// ---------------------------------------------------------------------------
// Muon multiple-scattering step, MI455X (gfx1250, wave32).
// Memory-bound elementwise kernel: 9 input streams, 4 output planes, plus a
// 40KB rad_length grid gathered per element. Grid is staged into LDS with the
// CDNA5 async global->LDS engine (GLOBAL_LOAD_ASYNC_TO_LDS_B128, ASYNCcnt);
// streams use b128 nontemporal loads/stores. Transcendentals use hardware
// TRANS ops (v_sin/v_cos/v_rcp/v_rsq) to keep the kernel firmly HBM-bound.
// ---------------------------------------------------------------------------

typedef float v4f __attribute__((ext_vector_type(4)));

#define GRID_DIM   100
#define GRID_ELEMS (GRID_DIM * GRID_DIM)   // 10000 floats
#define GRID_BYTES (GRID_ELEMS * 4)        // 40000 bytes
#define CHUNKS     (GRID_BYTES / 16)       // 2500 x b128 transfers
#define BLOCK      256                     // 8 wave32 waves
#define ITERS      8                       // float4 iterations per thread
// elements per block = BLOCK * 4 * ITERS = 8192

__global__ __launch_bounds__(BLOCK) void muon_scatter_kernel(
    const float* __restrict__ gx,   const float* __restrict__ gy,
    const float* __restrict__ gth,  const float* __restrict__ gtx,
    const float* __restrict__ gty,  const float* __restrict__ gmom,
    const float* __restrict__ grad, const float* __restrict__ gz1,
    const float* __restrict__ gz2,  const float* __restrict__ gpu,
    float* __restrict__ out, int n)
{
    __shared__ float s_grid[GRID_ELEMS];

    const int tid = threadIdx.x;

    // ---- Stage rad_length grid into LDS via CDNA5 async copy ------------
    {
        // The low 32 bits of a generic pointer to LDS are the LDS byte address.
        unsigned lds_base = (unsigned)(uintptr_t)(&s_grid[0]);
        for (int k = tid; k < CHUNKS; k += BLOCK) {
            unsigned loff = lds_base + (unsigned)(k * 16);
            unsigned goff = (unsigned)(k * 16);
            // GVS mode: copies 16 bytes from grad plus the offset in goff to the LDS address in loff
            asm volatile("global_load_async_to_lds_b128 %0, %1, %2"
                         :
                         : "v"(loff), "v"(goff), "s"(grad)
                         : "memory");
        }
#if __has_builtin(__builtin_amdgcn_s_wait_asynccnt)
        __builtin_amdgcn_s_wait_asynccnt(0);
#else
        asm volatile("s_wait_asynccnt 0" ::: "memory");
#endif
        __syncthreads();
    }

    // ---- Streaming compute ---------------------------------------------
    const v4f* X  = (const v4f*)gx;
    const v4f* Y  = (const v4f*)gy;
    const v4f* TH = (const v4f*)gth;
    const v4f* TX = (const v4f*)gtx;
    const v4f* TY = (const v4f*)gty;
    const v4f* PM = (const v4f*)gmom;
    const v4f* Z1 = (const v4f*)gz1;
    const v4f* Z2 = (const v4f*)gz2;
    const v4f* PU = (const v4f*)gpu;
    v4f*       O  = (v4f*)out;

    const int n4    = n >> 2;                 // float4 elements per stream
    const int base4 = blockIdx.x * (BLOCK * ITERS);

    const float DZ         = 0.1f;
    const float A_SQRTDZ   = 0.0043006974f;   // 0.0136 * sqrt(0.1)
    const float SQRT2      = 1.41421356237309515f;
    const float INV_SQRT12 = 0.28867513459481287f;
    const float TWO_PI     = 6.2831853071795865f;
    const float INV_SIZE   = 100.0f;

    for (int it = 0; it < ITERS; ++it) {
        const int i4 = base4 + it * BLOCK + tid;
        if (i4 >= n4) break;

        v4f vx  = __builtin_nontemporal_load(&X[i4]);
        v4f vy  = __builtin_nontemporal_load(&Y[i4]);
        v4f vth = __builtin_nontemporal_load(&TH[i4]);
        v4f vtx = __builtin_nontemporal_load(&TX[i4]);
        v4f vty = __builtin_nontemporal_load(&TY[i4]);
        v4f vpm = __builtin_nontemporal_load(&PM[i4]);
        v4f vz1 = __builtin_nontemporal_load(&Z1[i4]);
        v4f vz2 = __builtin_nontemporal_load(&Z2[i4]);
        v4f vpu = __builtin_nontemporal_load(&PU[i4]);

        v4f xo, yo, txo, tyo;

#pragma unroll
        for (int c = 0; c < 4; ++c) {
            const float x  = vx[c];
            const float y  = vy[c];
            const float th = vth[c];
            const float tx = vtx[c];
            const float ty = vty[c];
            const float pm = vpm[c];
            const float z1 = vz1[c];
            const float z2 = vz2[c];
            const float pu = vpu[c];

            const bool mask = (x >= 0.0f) & (x < 1.0f) & (y >= 0.0f) & (y < 1.0f);

            int ix = (int)floorf(x * INV_SIZE);
            int iy = (int)floorf(y * INV_SIZE);
            ix = ix < 0 ? 0 : (ix > GRID_DIM - 1 ? GRID_DIM - 1 : ix);
            iy = iy < 0 ? 0 : (iy > GRID_DIM - 1 ? GRID_DIM - 1 : iy);

            const float x0 = s_grid[ix * GRID_DIM + iy];   // ds_load_b32 gather

            // theta0 = (A/mom) * sqrt(DZ / (x0*cos(th)))
            //        = (A*sqrt(DZ)) * rcp(mom) * rsq(x0*cos(th))
            const float theta0 = A_SQRTDZ
                               * __builtin_amdgcn_rcpf(pm)
                               * __builtin_amdgcn_rsqf(x0 * __cosf(th));
            const float theta_msc = SQRT2 * z2 * theta0;

            const float phi  = pu * TWO_PI;
            const float cphi = __cosf(phi);
            const float sphi = __sinf(phi);

            const float dh = DZ * __sinf(theta0) * (z1 * INV_SQRT12 + 0.5f * z2);

            const float c_tx = __cosf(tx);
            const float c_ty = __cosf(ty);
            const float s_tx = __sinf(tx);
            const float s_ty = __sinf(ty);
            const float tan_tx = s_tx * __builtin_amdgcn_rcpf(c_tx);
            const float tan_ty = s_ty * __builtin_amdgcn_rcpf(c_ty);

            const float dx = SQRT2 * dh * cphi * c_tx;
            const float dy = SQRT2 * dh * sphi * c_ty;

            xo[c]  = (mask ? x + dx : x) + DZ * tan_tx;
            yo[c]  = (mask ? y + dy : y) + DZ * tan_ty;
            txo[c] = mask ? tx + theta_msc * cphi : tx;
            tyo[c] = mask ? ty + theta_msc * sphi : ty;
        }

        __builtin_nontemporal_store(xo,  &O[0 * n4 + i4]);
        __builtin_nontemporal_store(yo,  &O[1 * n4 + i4]);
        __builtin_nontemporal_store(txo, &O[2 * n4 + i4]);
        __builtin_nontemporal_store(tyo, &O[3 * n4 + i4]);
    }
}

extern "C" void kernel_launch(void* const* d_in, const int* in_sizes, int n_in,
                              void* d_out, int out_size, void* d_ws, size_t ws_size,
                              hipStream_t stream) {
    (void)n_in; (void)out_size; (void)d_ws; (void)ws_size;

    const float* x   = (const float*)d_in[0];
    const float* y   = (const float*)d_in[1];
    const float* th  = (const float*)d_in[2];
    const float* tx  = (const float*)d_in[3];
    const float* ty  = (const float*)d_in[4];
    const float* mom = (const float*)d_in[5];
    const float* rad = (const float*)d_in[6];
    const float* z1  = (const float*)d_in[7];
    const float* z2  = (const float*)d_in[8];
    const float* pu  = (const float*)d_in[9];
    float* out       = (float*)d_out;

    const int n  = in_sizes[0];          // N = 8388608 (divisible by 4)
    const int n4 = n / 4;
    const int per_block = BLOCK * ITERS; // float4 elements per block
    const int blocks = (n4 + per_block - 1) / per_block;

    muon_scatter_kernel<<<dim3(blocks), dim3(BLOCK), 0, stream>>>(
        x, y, th, tx, ty, mom, rad, z1, z2, pu, out, n);
}